// AdditiveAttention_11699490914902
// MI455X (gfx1250) — compile-verified
//
#include <hip/hip_runtime.h>

// Problem constants (match reference)
#define B_  4
#define Q_  512
#define K_  1024
#define QD_ 512
#define VD_ 512
#define H_  128

typedef __attribute__((ext_vector_type(2))) float v2f;
typedef __attribute__((ext_vector_type(8))) float v8f;

// Probe (device pass only): fail loudly if the gfx1250 hardware-tanh builtin
// is missing. Round-2 feedback confirmed the DEVICE pass passes this gate,
// i.e. __builtin_amdgcn_tanhf exists for gfx1250; only the host pass (which
// also defines __AMDGCN__ under this hipcc) tripped the old guard.
#if defined(__HIP_DEVICE_COMPILE__) && !__has_builtin(__builtin_amdgcn_tanhf)
#error "gfx1250: __builtin_amdgcn_tanhf not available - switch to inline-asm v_tanh_f32"
#endif

__device__ __forceinline__ float fast_tanh(float x) {
#if defined(__HIP_DEVICE_COMPILE__)
    return __builtin_amdgcn_tanhf(x);   // v_tanh_f32: TRANS pipe, co-executes with VALU
#else
    return tanhf(x);                    // host pass only (never executed)
#endif
}

// ---------------------------------------------------------------------------
// Generic fp32 WMMA GEMM: C[M x N] = A[M x K] * B[K x N], row-major, batched
// via blockIdx.y with element strides sA/sB/sC. Each wave computes a 16x64
// tile using V_WMMA_F32_16X16X4_F32 with 4 accumulators (A fragment reused 4x).
// Requires M%16==0, N%64==0, K%4==0; grid.x*8 waves == (M/16)*(N/64).
//
// ISA layouts (CDNA5 ISA §7.12.2, wave32):
//   A 16x4 : lane (0-15 / 16-31) -> M = lane&15 ; VGPR0/1 hold K = 2*(lane>>4) + {0,1}
//   B 4x16 : VGPR0/1 hold K-rows {0,1} (lanes 0-15) / {2,3} (lanes 16-31), N = lane&15
//   D 16x16: VGPR g -> row g + 8*(lane>>4), col = lane&15
// ---------------------------------------------------------------------------
__global__ __launch_bounds__(256) void wmma_gemm_f32(
    const float* __restrict__ A, const float* __restrict__ Bm, float* __restrict__ C,
    int M, int N, int Kd, int lda, int ldb, int ldc,
    long sA, long sB, long sC)
{
    const int tid    = threadIdx.x;
    const int lane   = tid & 31;
    const int wave   = tid >> 5;
    const int tilesN = N >> 6;                   // N / 64
    const int wt     = blockIdx.x * 8 + wave;
    const int tm     = wt / tilesN;
    const int tn     = wt % tilesN;
    if (tm * 16 >= M) return;

    const float* Ab = A  + (size_t)blockIdx.y * sA;
    const float* Bb = Bm + (size_t)blockIdx.y * sB;
    float*       Cb = C  + (size_t)blockIdx.y * sC;

    const int m     = lane & 15;                 // row-in-tile (A) / col-in-subtile (B,D)
    const int khalf = lane >> 4;                 // 0 or 1: K sub-pair selector
    const int row   = tm * 16 + m;
    const int col0  = tn * 64 + m;

    v8f acc0 = {}, acc1 = {}, acc2 = {}, acc3 = {};

    const float* arow = Ab + (size_t)row * lda + 2 * khalf;
    for (int k0 = 0; k0 < Kd; k0 += 4) {
        v2f a;
        a.x = arow[k0];
        a.y = arow[k0 + 1];                      // adjacent -> merges to b64 load

        const float* brow = Bb + (size_t)(k0 + 2 * khalf) * ldb + col0;
        v2f b0, b1, b2, b3;
        b0.x = brow[0];   b0.y = brow[0  + ldb];
        b1.x = brow[16];  b1.y = brow[16 + ldb];
        b2.x = brow[32];  b2.y = brow[32 + ldb];
        b3.x = brow[48];  b3.y = brow[48 + ldb];

        acc0 = __builtin_amdgcn_wmma_f32_16x16x4_f32(false, a, false, b0, (short)0, acc0, false, false);
        acc1 = __builtin_amdgcn_wmma_f32_16x16x4_f32(false, a, false, b1, (short)0, acc1, false, false);
        acc2 = __builtin_amdgcn_wmma_f32_16x16x4_f32(false, a, false, b2, (short)0, acc2, false, false);
        acc3 = __builtin_amdgcn_wmma_f32_16x16x4_f32(false, a, false, b3, (short)0, acc3, false, false);
    }

    #pragma unroll
    for (int g = 0; g < 8; ++g) {
        const int r = tm * 16 + g + 8 * khalf;
        float* crow = Cb + (size_t)r * ldc + tn * 64 + m;
        crow[0]  = acc0[g];
        crow[16] = acc1[g];
        crow[32] = acc2[g];
        crow[48] = acc3[g];
    }
}

// ---------------------------------------------------------------------------
// scores[b,q,k] = sum_h v[h] * tanh(qp[b,q,h] + kp[b,k,h]); softmax over k.
// One block (256 threads) per (b,q); each thread owns 4 k-rows.
// qp-row and v staged in LDS; kp streamed with float4 loads.
// ---------------------------------------------------------------------------
__global__ __launch_bounds__(256) void scores_softmax_kernel(
    const float* __restrict__ qp, const float* __restrict__ kp,
    const float* __restrict__ v, float* __restrict__ attn)
{
    __shared__ float qs[H_];
    __shared__ float vs[H_];
    __shared__ float red[8];

    const int tid = threadIdx.x;
    const int bq  = blockIdx.x;       // b*Q + q
    const int b   = bq >> 9;          // Q_ == 512

    if (tid < H_) {
        qs[tid] = qp[(size_t)bq * H_ + tid];
        vs[tid] = v[tid];
    }
    __syncthreads();

    float s[4];
    #pragma unroll
    for (int j = 0; j < 4; ++j) {
        const int k = tid + j * 256;
        const float4* kr = reinterpret_cast<const float4*>(kp + ((size_t)b * K_ + k) * H_);
        float acc = 0.f;
        #pragma unroll 8
        for (int h4 = 0; h4 < H_ / 4; ++h4) {
            float4 kk = kr[h4];
            acc += vs[4*h4+0] * fast_tanh(qs[4*h4+0] + kk.x);
            acc += vs[4*h4+1] * fast_tanh(qs[4*h4+1] + kk.y);
            acc += vs[4*h4+2] * fast_tanh(qs[4*h4+2] + kk.z);
            acc += vs[4*h4+3] * fast_tanh(qs[4*h4+3] + kk.w);
        }
        s[j] = acc;
    }

    // ---- block max over 1024 scores ----
    float mx = fmaxf(fmaxf(s[0], s[1]), fmaxf(s[2], s[3]));
    #pragma unroll
    for (int o = 16; o > 0; o >>= 1) mx = fmaxf(mx, __shfl_xor(mx, o, 32));
    if ((tid & 31) == 0) red[tid >> 5] = mx;
    __syncthreads();
    mx = red[0];
    #pragma unroll
    for (int w = 1; w < 8; ++w) mx = fmaxf(mx, red[w]);
    __syncthreads();                   // red[] reused below

    // ---- exp + block sum ----
    float e[4];
    float sum = 0.f;
    #pragma unroll
    for (int j = 0; j < 4; ++j) { e[j] = __expf(s[j] - mx); sum += e[j]; }
    #pragma unroll
    for (int o = 16; o > 0; o >>= 1) sum += __shfl_xor(sum, o, 32);
    if ((tid & 31) == 0) red[tid >> 5] = sum;
    __syncthreads();
    sum = 0.f;
    #pragma unroll
    for (int w = 0; w < 8; ++w) sum += red[w];

    const float inv = 1.0f / sum;
    #pragma unroll
    for (int j = 0; j < 4; ++j)
        attn[(size_t)bq * K_ + tid + j * 256] = e[j] * inv;
}

// ---------------------------------------------------------------------------
extern "C" void kernel_launch(void* const* d_in, const int* in_sizes, int n_in,
                              void* d_out, int out_size, void* d_ws, size_t ws_size,
                              hipStream_t stream) {
    (void)in_sizes; (void)n_in; (void)out_size; (void)ws_size;

    const float* query = (const float*)d_in[0];   // [B,Q,QD]
    const float* key   = (const float*)d_in[1];   // [B,K,KD]
    const float* value = (const float*)d_in[2];   // [B,K,VD]
    const float* Wq    = (const float*)d_in[3];   // [QD,H]
    const float* Wk    = (const float*)d_in[4];   // [KD,H]
    const float* v     = (const float*)d_in[5];   // [H]

    float* ctx  = (float*)d_out;                              // [B,Q,VD]
    float* attn = (float*)d_out + (size_t)B_ * Q_ * VD_;      // [B,Q,K]

    float* qp = (float*)d_ws;                                 // [B*Q, H]  (1 MB)
    float* kpp = qp + (size_t)B_ * Q_ * H_;                   // [B*K, H]  (2 MB)

    // qp = query @ Wq : M=2048, N=128, K=512 -> (128*2)/8 = 32 blocks
    wmma_gemm_f32<<<dim3(32, 1), 256, 0, stream>>>(query, Wq, qp,
        B_ * Q_, H_, QD_, QD_, H_, H_, 0L, 0L, 0L);

    // kp = key @ Wk : M=4096 -> 64 blocks
    wmma_gemm_f32<<<dim3(64, 1), 256, 0, stream>>>(key, Wk, kpp,
        B_ * K_, H_, QD_, QD_, H_, H_, 0L, 0L, 0L);

    // scores + softmax -> attention (written directly to its d_out slot)
    scores_softmax_kernel<<<B_ * Q_, 256, 0, stream>>>(qp, kpp, v, attn);

    // context = attention @ value, batched over B: M=512,N=512,K=1024 -> 32 x 4 blocks
    wmma_gemm_f32<<<dim3(32, 4), 256, 0, stream>>>(attn, value, ctx,
        Q_, VD_, K_, K_, VD_, VD_,
        (long)Q_ * K_, (long)K_ * VD_, (long)Q_ * VD_);
}